// Encoder_17978733101771
// MI455X (gfx1250) — compile-verified
//
#include <hip/hip_runtime.h>
#include <hip/hip_bf16.h>

typedef float v2f __attribute__((ext_vector_type(2)));
typedef float v8f __attribute__((ext_vector_type(8)));

#define NNODES 10000
#define NEDGES 160000
#define KSTACK 4

// ---------------------------------------------------------------- utilities

__global__ void zero_f32(float* __restrict__ p, long long n) {
    long long i = (long long)blockIdx.x * blockDim.x + threadIdx.x;
    if (i < n) p[i] = 0.0f;
}

__global__ void deg_kernel(const int* __restrict__ ei, float* __restrict__ deg, int E) {
    int e = blockIdx.x * blockDim.x + threadIdx.x;
    if (e < E) atomicAdd(&deg[ei[2 * e + 1]], 1.0f);
}

__global__ void dis_kernel(const float* __restrict__ deg, float* __restrict__ dis, int N) {
    int n = blockIdx.x * blockDim.x + threadIdx.x;
    if (n < N) {
        float d = deg[n];
        dis[n] = (d > 0.0f) ? rsqrtf(fmaxf(d, 1.0f)) : 0.0f;
    }
}

__global__ void norm_kernel(const int* __restrict__ ei, const float* __restrict__ dis,
                            float* __restrict__ nrm, int E) {
    int e = blockIdx.x * blockDim.x + threadIdx.x;
    if (e < E) nrm[e] = dis[ei[2 * e]] * dis[ei[2 * e + 1]];
}

// edge_agg[dst] += norm_e * (edge_attr[e] @ ew + eb)   (one wave per edge)
__global__ void edge_emb_agg(const float* __restrict__ eattr, const int* __restrict__ ei,
                             const float* __restrict__ nrm, const float* __restrict__ ew,
                             const float* __restrict__ eb, float* __restrict__ out,
                             int E, int F) {
    int e = blockIdx.x * blockDim.y + threadIdx.y;
    if (e >= E) return;
    int lane = threadIdx.x;
    int dst = ei[2 * e + 1];
    float sc = nrm[e];
    const float4* ar = (const float4*)(eattr + (long long)e * 16);
    float4 a0 = ar[0], a1 = ar[1], a2 = ar[2], a3 = ar[3];
    float att[16] = {a0.x, a0.y, a0.z, a0.w, a1.x, a1.y, a1.z, a1.w,
                     a2.x, a2.y, a2.z, a2.w, a3.x, a3.y, a3.z, a3.w};
    int fpl = F >> 5;
    for (int j = 0; j < fpl; ++j) {
        int f = lane + j * 32;
        float s = eb[f];
#pragma unroll
        for (int d = 0; d < 16; ++d) s += att[d] * ew[d * F + f];
        atomicAdd(out + (long long)dst * F + f, s * sc);
    }
}

// dst[k][:] = src[:] replicated over k (float4 vectorized)
__global__ void init_rep(const float* __restrict__ s, float* __restrict__ d,
                         long long nf4, long long nf) {
    long long i = (long long)blockIdx.x * blockDim.x + threadIdx.x;
    if (i >= nf4) return;
    float4 v = ((const float4*)s)[i];
#pragma unroll
    for (int k = 0; k < KSTACK; ++k) ((float4*)(d + k * nf))[i] = v;
}

// dstF[k][dst] += norm_e * srcF[k][src]   (one wave per edge, float4 gathers)
__global__ void scatter_feat(const float* __restrict__ src, float* __restrict__ dst,
                             const int* __restrict__ ei, const float* __restrict__ nrm,
                             int E, int F, long long nf) {
    int e = blockIdx.x * blockDim.y + threadIdx.y;
    if (e >= E) return;
    int lane = threadIdx.x;
    int s = ei[2 * e], d = ei[2 * e + 1];
    float sc = nrm[e];
    int iters = F >> 7;  // F/128 : each lane does 4 floats per iter
#pragma unroll
    for (int k = 0; k < KSTACK; ++k) {
        const float4* srow = (const float4*)(src + k * nf + (long long)s * F);
        float* drow = dst + k * nf + (long long)d * F;
        for (int it = 0; it < iters; ++it) {
            float4 v = srow[lane + it * 32];
            int base = (lane + it * 32) * 4;
            atomicAdd(drow + base + 0, v.x * sc);
            atomicAdd(drow + base + 1, v.y * sc);
            atomicAdd(drow + base + 2, v.z * sc);
            atomicAdd(drow + base + 3, v.w * sc);
        }
    }
}

// ---------------------------------------------------------------- WMMA GEMM
// D[k] = (ADD ? D[k] : 0) + A[k] @ B[k] + (BIAS ? bias[k] : 0)
// one wave per 16x16 tile of D, K-dim in steps of 4 via V_WMMA_F32_16X16X4_F32.
template <bool ADD, bool BIAS>
__global__ void gemm_wmma(const float* __restrict__ A, long long aK, int lda,
                          const float* __restrict__ B, long long bK, int ldb,
                          const float* __restrict__ bias, int biasK,
                          float* __restrict__ D, long long dK, int ldd,
                          int Kdim, int rowTiles) {
    int tm = blockIdx.x * blockDim.y + threadIdx.y;
    if (tm >= rowTiles) return;  // wave-uniform: EXEC stays all-ones for live waves
    int lane = threadIdx.x;
    int hh = lane >> 4;      // half-wave select
    int l = lane & 15;
    int k = blockIdx.z;
    const float* Ak = A + (long long)k * aK + (long long)(tm * 16) * lda;
    const float* Bk = B + (long long)k * bK + blockIdx.y * 16;
    v8f acc = {};
    for (int k0 = 0; k0 < Kdim; k0 += 4) {
        v2f a, b;
        // A 16x4 layout: VGPR0 lanes0-15 K=k0, lanes16-31 K=k0+2; VGPR1 = +1
        a.x = Ak[(long long)l * lda + k0 + 2 * hh];
        a.y = Ak[(long long)l * lda + k0 + 2 * hh + 1];
        // B 4x16 layout: VGPR0 lanes0-15 row k0, lanes16-31 row k0+2; VGPR1 = +1
        b.x = Bk[(long long)(k0 + 2 * hh) * ldb + l];
        b.y = Bk[(long long)(k0 + 2 * hh + 1) * ldb + l];
        acc = __builtin_amdgcn_wmma_f32_16x16x4_f32(false, a, false, b,
                                                    (short)0, acc, false, false);
    }
    float* Dk = D + (long long)k * dK + (long long)(tm * 16) * ldd + blockIdx.y * 16;
#pragma unroll
    for (int v = 0; v < 8; ++v) {
        int row = v + 8 * hh;  // C layout: VGPR v -> rows v and v+8
        float o = acc[v];
        if (BIAS) o += bias[k * biasK + blockIdx.y * 16 + l];
        if (ADD) o += Dk[(long long)row * ldd + l];
        Dk[(long long)row * ldd + l] = o;
    }
}

// ------------------------------------------------- mean(K) + GroupNorm + tanh
template <int CPG>
__global__ void gn_tanh(const float* __restrict__ Q, long long nf,
                        const float* __restrict__ gamma, const float* __restrict__ beta,
                        float* __restrict__ out, int N, int C) {
    int t = blockIdx.x * blockDim.x + threadIdx.x;
    if (t >= N * 16) return;
    int n = t >> 4, g = t & 15;
    int c0 = g * CPG;
    float vals[CPG];
    float mu = 0.0f;
#pragma unroll
    for (int c = 0; c < CPG; ++c) {
        long long idx = (long long)n * C + c0 + c;
        float v = 0.25f * (Q[idx] + Q[idx + nf] + Q[idx + 2 * nf] + Q[idx + 3 * nf]);
        vals[c] = v;
        mu += v;
    }
    mu *= (1.0f / CPG);
    float var = 0.0f;
#pragma unroll
    for (int c = 0; c < CPG; ++c) {
        float d = vals[c] - mu;
        var += d * d;
    }
    var *= (1.0f / CPG);
    float inv = rsqrtf(var + 1e-5f);
#pragma unroll
    for (int c = 0; c < CPG; ++c) {
        float xn = (vals[c] - mu) * inv;
        out[(long long)n * C + c0 + c] = tanhf(xn * gamma[c0 + c] + beta[c0 + c]);
    }
}

// ---------------------------------------------------------------- launcher

extern "C" void kernel_launch(void* const* d_in, const int* in_sizes, int n_in,
                              void* d_out, int out_size, void* d_ws, size_t ws_size,
                              hipStream_t stream) {
    const float* x = (const float*)d_in[0];
    const float* eattr = (const float*)d_in[1];
    const int* ei = (const int*)d_in[2];
    const float* ew1 = (const float*)d_in[3];
    const float* eb1 = (const float*)d_in[4];
    const float* iw1 = (const float*)d_in[5];
    const float* w1 = (const float*)d_in[6];
    const float* rw1 = (const float*)d_in[7];
    const float* b1 = (const float*)d_in[8];
    const float* g1g = (const float*)d_in[9];
    const float* g1b = (const float*)d_in[10];
    const float* ew2 = (const float*)d_in[11];
    const float* eb2 = (const float*)d_in[12];
    const float* iw2 = (const float*)d_in[13];
    const float* w2 = (const float*)d_in[14];
    const float* rw2 = (const float*)d_in[15];
    const float* b2 = (const float*)d_in[16];
    const float* g2g = (const float*)d_in[17];
    const float* g2b = (const float*)d_in[18];

    const int N = NNODES, E = NEDGES, K = KSTACK;
    const long long NF1 = (long long)N * 128, NF2 = (long long)N * 256;

    float* ws = (float*)d_ws;
    float* deg = ws;
    float* dis = deg + N;
    float* nrm = dis + N;
    float* ea1 = nrm + E;
    float* ea2 = ea1 + NF1;
    float* h1 = ea2 + NF2;
    float* P = h1 + NF1;
    float* Qb = P + K * NF2;

    dim3 wblk(32, 4);
    int egrid = (E + 3) / 4;
    int rt = N / 16;  // 625 row tiles, N divisible by 16
    dim3 gg1((rt + 3) / 4, 128 / 16, K);
    dim3 gg2((rt + 3) / 4, 256 / 16, K);

    // ---- norm + edge aggregates (once per call)
    zero_f32<<<(N + 255) / 256, 256, 0, stream>>>(deg, N);
    zero_f32<<<(int)((NF1 + NF2 + 255) / 256), 256, 0, stream>>>(ea1, NF1 + NF2);
    deg_kernel<<<(E + 255) / 256, 256, 0, stream>>>(ei, deg, E);
    dis_kernel<<<(N + 255) / 256, 256, 0, stream>>>(deg, dis, N);
    norm_kernel<<<(E + 255) / 256, 256, 0, stream>>>(ei, dis, nrm, E);
    edge_emb_agg<<<egrid, wblk, 0, stream>>>(eattr, ei, nrm, ew1, eb1, ea1, E, 128);
    edge_emb_agg<<<egrid, wblk, 0, stream>>>(eattr, ei, nrm, ew2, eb2, ea2, E, 256);

    // ---- layer 1 (F_IN=64 -> MID=128)
    gemm_wmma<false, false><<<gg1, wblk, 0, stream>>>(
        x, 0LL, 64, iw1, (long long)64 * 128, 128, nullptr, 0, P, NF1, 128, 64, rt);
    // t = 0
    init_rep<<<(int)((NF1 / 4 + 255) / 256), 256, 0, stream>>>(ea1, Qb, NF1 / 4, NF1);
    scatter_feat<<<egrid, wblk, 0, stream>>>(P, Qb, ei, nrm, E, 128, NF1);
    gemm_wmma<true, true><<<gg1, wblk, 0, stream>>>(
        x, 0LL, 64, rw1, (long long)64 * 128, 128, b1, 128, Qb, NF1, 128, 64, rt);
    // t = 1
    gemm_wmma<false, false><<<gg1, wblk, 0, stream>>>(
        Qb, NF1, 128, w1, (long long)128 * 128, 128, nullptr, 0, P, NF1, 128, 128, rt);
    init_rep<<<(int)((NF1 / 4 + 255) / 256), 256, 0, stream>>>(ea1, Qb, NF1 / 4, NF1);
    scatter_feat<<<egrid, wblk, 0, stream>>>(P, Qb, ei, nrm, E, 128, NF1);
    gemm_wmma<true, true><<<gg1, wblk, 0, stream>>>(
        x, 0LL, 64, rw1 + (long long)K * 64 * 128, (long long)64 * 128, 128,
        b1 + K * 128, 128, Qb, NF1, 128, 64, rt);
    gn_tanh<8><<<(N * 16 + 255) / 256, 256, 0, stream>>>(Qb, NF1, g1g, g1b, h1, N, 128);

    // ---- layer 2 (MID=128 -> OUT=256)
    gemm_wmma<false, false><<<gg2, wblk, 0, stream>>>(
        h1, 0LL, 128, iw2, (long long)128 * 256, 256, nullptr, 0, P, NF2, 256, 128, rt);
    // t = 0
    init_rep<<<(int)((NF2 / 4 + 255) / 256), 256, 0, stream>>>(ea2, Qb, NF2 / 4, NF2);
    scatter_feat<<<egrid, wblk, 0, stream>>>(P, Qb, ei, nrm, E, 256, NF2);
    gemm_wmma<true, true><<<gg2, wblk, 0, stream>>>(
        h1, 0LL, 128, rw2, (long long)128 * 256, 256, b2, 256, Qb, NF2, 256, 128, rt);
    // t = 1
    gemm_wmma<false, false><<<gg2, wblk, 0, stream>>>(
        Qb, NF2, 256, w2, (long long)256 * 256, 256, nullptr, 0, P, NF2, 256, 256, rt);
    init_rep<<<(int)((NF2 / 4 + 255) / 256), 256, 0, stream>>>(ea2, Qb, NF2 / 4, NF2);
    scatter_feat<<<egrid, wblk, 0, stream>>>(P, Qb, ei, nrm, E, 256, NF2);
    gemm_wmma<true, true><<<gg2, wblk, 0, stream>>>(
        h1, 0LL, 128, rw2 + (long long)K * 128 * 256, (long long)128 * 256, 256,
        b2 + K * 256, 256, Qb, NF2, 256, 128, rt);
    gn_tanh<16><<<(N * 16 + 255) / 256, 256, 0, stream>>>(Qb, NF2, g2g, g2b,
                                                          (float*)d_out, N, 256);
}